// OpeningLoss2D_61349312856565
// MI455X (gfx1250) — compile-verified
//
#include <hip/hip_runtime.h>

// CDNA5 / gfx1250 — grey-opening (2x2 flat SE) MSE loss, bandwidth-bound.
// Input: labels f32 [8,16,512,512]. Output: scalar f32 mean((x - open(x))^2).

typedef __attribute__((ext_vector_type(2))) float v2f;
typedef __attribute__((ext_vector_type(8))) float v8f;

#define IMG_H 512
#define IMG_W 512
#define N_IMG 128            // 8 * 16 channels-as-images
#define ROWS_PER_BLOCK 64    // 8 strips per image -> 1024 blocks

// Wave-level reduction of 64 partials (2 per lane) with one
// V_WMMA_F32_16X16X4_F32:  D = A(16x4) * ones(4x16).
// A layout (32-bit, 16x4): lanes 0-15 supply K=0 (a.x) and K=1 (a.y) for
// M=lane; lanes 16-31 supply K=2/K=3 for M=lane-16.  With B == 1 every
// D[m][n] equals rowsum_m, so the sum of a lane's 8 D components is
// sum(rows 0..7) on lanes 0-15 and sum(rows 8..15) on lanes 16-31; one
// xor-16 shuffle completes the wave total.
static __device__ __forceinline__ float wave_wmma_reduce(float acc0, float acc1) {
  v2f a; a.x = acc0; a.y = acc1;
  v2f b; b.x = 1.0f; b.y = 1.0f;
  v8f c = {};
  v8f d = __builtin_amdgcn_wmma_f32_16x16x4_f32(
      /*neg_a=*/false, a, /*neg_b=*/false, b,
      /*c_mod=*/(short)0, c, /*reuse_a=*/false, /*reuse_b=*/false);
  float s = d[0] + d[1] + d[2] + d[3] + d[4] + d[5] + d[6] + d[7];
  s += __shfl_xor(s, 16, 32);
  return s;
}

template <bool ATOMIC>
__global__ __launch_bounds__(256) void opening_mse_kernel(
    const float* __restrict__ x, float* __restrict__ out) {
  const int t    = threadIdx.x;
  const int colq = t & 127;     // column quad: covers columns 4*colq .. 4*colq+3
  const int rgrp = t >> 7;      // 0..1, interleaved rows
  const int img   = blockIdx.x >> 3;
  const int strip = blockIdx.x & 7;
  const float* __restrict__ base = x + (size_t)img * (IMG_H * IMG_W);

  const int c0 = colq << 2;
  const int cl = (c0 == 0) ? 0 : c0 - 1;                       // left halo col (clamped)
  const bool rclamp = (c0 + 4 > IMG_W - 1);                    // only colq==127
  const int cr = rclamp ? (IMG_W - 1) : (c0 + 4);              // right halo col

  float acc0 = 0.0f, acc1 = 0.0f;

  for (int k = 0; k < ROWS_PER_BLOCK / 2; ++k) {
    const int i   = strip * ROWS_PER_BLOCK + rgrp + 2 * k;
    const int im1 = (i == 0) ? 0 : i - 1;
    const int ip1 = (i == IMG_H - 1) ? i : i + 1;
    const float* __restrict__ rm = base + (size_t)im1 * IMG_W;
    const float* __restrict__ rc = base + (size_t)i   * IMG_W;
    const float* __restrict__ rp = base + (size_t)ip1 * IMG_W;

    // Prefetch a couple of rows ahead (clamped, stays in-buffer).
    const int ipf = (i + 2 > IMG_H - 1) ? (IMG_H - 1) : (i + 2);
    __builtin_prefetch(base + (size_t)ipf * IMG_W + c0, 0, 0);

    // Coalesced main loads (512B per wave per row) + scalar halos (cache hits).
    float4 a = *(const float4*)(rm + c0);
    float4 b = *(const float4*)(rc + c0);
    float4 c = *(const float4*)(rp + c0);
    float aL = rm[cl], bL = rc[cl], cL = rp[cl];
    float aR = rm[cr], bR = rc[cr], cR = rp[cr];

    // Horizontal erosion m[q] = min(x[j-1], x[j]) for j = c0+q, q=0..4.
    float mA0 = fminf(aL, a.x), mA1 = fminf(a.x, a.y), mA2 = fminf(a.y, a.z),
          mA3 = fminf(a.z, a.w), mA4 = fminf(a.w, aR);
    float mB0 = fminf(bL, b.x), mB1 = fminf(b.x, b.y), mB2 = fminf(b.y, b.z),
          mB3 = fminf(b.z, b.w), mB4 = fminf(b.w, bR);
    float mC0 = fminf(cL, c.x), mC1 = fminf(c.x, c.y), mC2 = fminf(c.y, c.z),
          mC3 = fminf(c.z, c.w), mC4 = fminf(c.w, cR);

    // Vertical erosion: eA = er(i, j) (rows i-1,i); eB = er(i+1, j) (rows i,i+1).
    float eA0 = fminf(mA0, mB0), eA1 = fminf(mA1, mB1), eA2 = fminf(mA2, mB2),
          eA3 = fminf(mA3, mB3), eA4 = fminf(mA4, mB4);
    float eB0 = fminf(mB0, mC0), eB1 = fminf(mB1, mC1), eB2 = fminf(mB2, mC2),
          eB3 = fminf(mB3, mC3), eB4 = fminf(mB4, mC4);
    // Dilation column clamp: er(., min(j+1,511)) == er(., 511) at the image edge.
    if (rclamp) { eA4 = eA3; eB4 = eB3; }

    // dil(i,j) = max(er(i,j), er(i,j+1), er(i+1,j), er(i+1,j+1)); MSE accumulate.
    float d0 = b.x - fmaxf(fmaxf(eA0, eA1), fmaxf(eB0, eB1));
    float d1 = b.y - fmaxf(fmaxf(eA1, eA2), fmaxf(eB1, eB2));
    float d2 = b.z - fmaxf(fmaxf(eA2, eA3), fmaxf(eB2, eB3));
    float d3 = b.w - fmaxf(fmaxf(eA3, eA4), fmaxf(eB3, eB4));
    acc0 = fmaf(d0, d0, acc0);
    acc1 = fmaf(d1, d1, acc1);
    acc0 = fmaf(d2, d2, acc0);
    acc1 = fmaf(d3, d3, acc1);
  }

  // Wave reduction via WMMA (EXEC all-ones here: full 256-thread block,
  // uniform control flow at this point).
  float wave_sum = wave_wmma_reduce(acc0, acc1);

  __shared__ float wsum[8];
  const int wid = t >> 5;
  if ((t & 31) == 0) wsum[wid] = wave_sum;
  __syncthreads();
  if (t == 0) {
    float s = wsum[0];
#pragma unroll
    for (int w = 1; w < 8; ++w) s += wsum[w];
    if (ATOMIC) atomicAdd(out, s);
    else        out[blockIdx.x] = s;
  }
}

__global__ __launch_bounds__(256) void reduce_partials_kernel(
    const float* __restrict__ part, float* __restrict__ out, int n, float scale) {
  __shared__ float sm[256];
  float s = 0.0f;
  for (int i = threadIdx.x; i < n; i += 256) s += part[i];
  sm[threadIdx.x] = s;
  __syncthreads();
  for (int off = 128; off > 0; off >>= 1) {
    if (threadIdx.x < off) sm[threadIdx.x] += sm[threadIdx.x + off];
    __syncthreads();
  }
  if (threadIdx.x == 0) out[0] = sm[0] * scale;
}

__global__ void zero_kernel(float* p) { p[0] = 0.0f; }
__global__ void scale_kernel(float* p, float scale) { p[0] *= scale; }

extern "C" void kernel_launch(void* const* d_in, const int* in_sizes, int n_in,
                              void* d_out, int out_size, void* d_ws, size_t ws_size,
                              hipStream_t stream) {
  const float* x = (const float*)d_in[0];
  float* out = (float*)d_out;
  const int nblocks = N_IMG * (IMG_H / ROWS_PER_BLOCK);  // 1024
  const float scale = 1.0f / (float)in_sizes[0];         // 1 / 33554432

  if (ws_size >= (size_t)nblocks * sizeof(float)) {
    // Fully deterministic two-pass reduction through workspace partials.
    float* part = (float*)d_ws;
    opening_mse_kernel<false><<<nblocks, 256, 0, stream>>>(x, part);
    reduce_partials_kernel<<<1, 256, 0, stream>>>(part, out, nblocks, scale);
  } else {
    // Fallback: accumulate directly into d_out with atomics.
    zero_kernel<<<1, 1, 0, stream>>>(out);
    opening_mse_kernel<true><<<nblocks, 256, 0, stream>>>(x, out);
    scale_kernel<<<1, 1, 0, stream>>>(out, scale);
  }
}